// ROIPooling_24515673325871
// MI455X (gfx1250) — compile-verified
//
#include <hip/hip_runtime.h>

// Problem constants (match reference)
#define B_     4
#define R_     160
#define H_     128
#define W_     128
#define C_     64
#define P_     7
#define NREG_  32
#define IOU_THR_ 0.4f
#define MAXBIN_ 14   // max bin extent: max over h in [7,63] of h - 6*(h/7) == 14 (roi wh < 64)

// ---------------------------------------------------------------------------
// _fix_span, matching the jnp.where cascade exactly.
// Only pad>0 branches mutate; for pad<=0 the C-vs-Python floordiv mismatch is
// irrelevant (lo >= 0 >= pad/2 and s-hi >= 0 > (1+pad)/2 in both semantics).
// ---------------------------------------------------------------------------
__device__ __forceinline__ void fix_span(int& lo, int& hi, int p, int s) {
  int pad = p - (hi - lo);
  bool pos  = pad > 0;
  bool fmin = lo < (pad / 2);
  bool fmax = (s - hi) < ((1 + pad) / 2);
  bool sym  = pos && !(fmin || fmax);
  int lo2 = lo, hi2 = hi;
  if (sym)         { lo2 = lo - pad / 2; hi2 = hi + (1 + pad) / 2; }
  if (pos && fmin) { lo2 = 0;            hi2 = p; }
  if (pos && fmax) { lo2 = s - p;        hi2 = s; }   // fmax overrides fmin, like the where() order
  lo = lo2; hi = hi2;
}

// ---------------------------------------------------------------------------
// Kernel 1: greedy NMS (sequential suppression scan, as in the reference
// lax.scan) + select first 32 survivors (default idx = R-1) + clip/fix-span.
// One block per batch image, 160 threads (5 wave32 waves), state in LDS.
// Thread 0 does only the inherently-serial rank compaction; threads 0..31
// then clip one region each. Writes int4 {x0,y0,w,h} per (b,region) to ws.
// ---------------------------------------------------------------------------
__global__ __launch_bounds__(R_) void nms_clip_kernel(const float* __restrict__ roi,
                                                      int* __restrict__ boxes_out) {
  __shared__ float bx[R_], by[R_], bw[R_], bh[R_], ba[R_];
  __shared__ int keep[R_];
  __shared__ int sel[NREG_];

  const int b = blockIdx.x;
  const int j = threadIdx.x;

  const float* r = roi + (b * R_ + j) * 4;
  bx[j] = r[0]; by[j] = r[1]; bw[j] = r[2]; bh[j] = r[3];
  ba[j] = r[2] * r[3];
  keep[j] = 1;

  // step i: sup = keep[i] & (iou[i,:] > thr) & (ar > i); keep &= ~sup
  for (int i = 0; i < R_ - 1; ++i) {
    __syncthreads();                 // publish prior iteration's writes
    if (keep[i] && j > i) {          // only lanes j>i write keep[j]; no race with keep[i]
      float xa = fmaxf(bx[i], bx[j]);
      float ya = fmaxf(by[i], by[j]);
      float xb = fminf(bx[i] + bw[i], bx[j] + bw[j]);
      float yb = fminf(by[i] + bh[i], by[j] + bh[j]);
      float inter = fmaxf(xb - xa, 0.0f) * fmaxf(yb - ya, 0.0f);
      float iou = inter / (ba[i] + ba[j] - inter);
      if (iou > IOU_THR_) keep[j] = 0;
    }
  }
  __syncthreads();

  if (j == 0) {                       // serial rank compaction (inherently sequential)
    int cnt = 0;
    for (int q = 0; q < R_ && cnt < NREG_; ++q)
      if (keep[q]) sel[cnt++] = q;    // rank < 32 kept; rank >= 32 dropped
    for (; cnt < NREG_; ++cnt) sel[cnt] = R_ - 1;   // default = last box
  }
  __syncthreads();

  if (j < NREG_) {                    // parallel clip + fix_span + store
    int s = sel[j];
    float fx = bx[s], fy = by[s], fw = bw[s], fh = bh[s];
    // astype(int32) truncation; coords are non-negative
    int x0 = max(0, (int)fx);
    int y0 = max(0, (int)fy);
    int x1 = min(W_, (int)(fx + fw));
    int y1 = min(H_, (int)(fy + fh));
    fix_span(x0, x1, P_, W_);
    fix_span(y0, y1, P_, H_);
    int* o = boxes_out + (b * NREG_ + j) * 4;
    o[0] = x0; o[1] = y0; o[2] = x1 - x0; o[3] = y1 - y0;
  }
}

// ---------------------------------------------------------------------------
// Kernel 2: 7x7 max pooling. One block per (b, region, ph, pw) bin,
// 64 threads = 2 wave32 waves, thread == channel.
//
// Pipeline (pure same-wave dataflow, no workgroup barrier needed):
//   wave lanes c issue one gfx1250 async Global->LDS dword copy per (row,col)
//   position (each instruction moves that wave's 32 channels, 128B).
//   Async loads retire IN ORDER (ISA 08 §4.1), so after issuing n copies:
//     s_wait_asynccnt 32  => first n-32 positions resident  -> reduce them
//     s_wait_asynccnt 0   => all resident                   -> reduce tail
//   overlapping the ds_load/v_max reduction with the in-flight tail copies.
//   LDS reads are bank-conflict free (dword index % 64 == c for all accesses).
// ---------------------------------------------------------------------------
__device__ __forceinline__ void wait_asynccnt_32() {
#if defined(__has_builtin) && __has_builtin(__builtin_amdgcn_s_wait_asynccnt)
  __builtin_amdgcn_s_wait_asynccnt(32);
#else
  asm volatile("s_wait_asynccnt 32" ::: "memory");
#endif
}
__device__ __forceinline__ void wait_asynccnt_0() {
#if defined(__has_builtin) && __has_builtin(__builtin_amdgcn_s_wait_asynccnt)
  __builtin_amdgcn_s_wait_asynccnt(0);
#else
  asm volatile("s_wait_asynccnt 0" ::: "memory");
#endif
}

__global__ __launch_bounds__(C_) void roi_pool_kernel(const float* __restrict__ feat,
                                                      const int* __restrict__ boxes,
                                                      float* __restrict__ out) {
  __shared__ float tile[MAXBIN_ * MAXBIN_ * C_];   // 14*14*64*4B = 49 KB of 320 KB WGP LDS

  int bid = blockIdx.x;
  const int pw = bid % P_;    bid /= P_;
  const int ph = bid % P_;    bid /= P_;
  const int k  = bid % NREG_; bid /= NREG_;
  const int b  = bid;
  const int c  = threadIdx.x;

  const int* bb = boxes + (b * NREG_ + k) * 4;
  const int x0 = bb[0], y0 = bb[1], w = bb[2], h = bb[3];
  const int hs  = h / P_;
  const int wsz = w / P_;
  const int r0 = y0 + ph * hs;
  const int r1 = (ph == P_ - 1) ? (y0 + h) : (r0 + hs);
  const int xa = x0 + pw * wsz;
  const int xb = (pw == P_ - 1) ? (x0 + w) : (xa + wsz);
  const int nr = min(r1 - r0, MAXBIN_);   // clamps are safety only; analysis bounds <= 14
  const int nc = min(xb - xa, MAXBIN_);

  // Raw LDS byte offset = low 32 bits of the generic address of a shared object
  // (flat->LDS mapping truncates to addr[31:0], ISA 00 §10.2).
  const unsigned lds_lane = (unsigned)(unsigned long long)(&tile[0]) + (unsigned)c * 4u;
  const float* gbase = feat + (((long long)b * H_ + r0) * W_ + xa) * C_ + c;

  // Issue all async copies; HW stalls issue if the 6-bit ASYNCcnt would
  // overflow, so no software throttling is needed.
  int idx = 0;
  for (int rr = 0; rr < nr; ++rr) {
    const float* grow = gbase + (long long)rr * (W_ * C_);
    for (int cc = 0; cc < nc; ++cc, ++idx) {
      unsigned dst = lds_lane + (unsigned)(idx * C_) * 4u;
      unsigned long long src = (unsigned long long)(grow + cc * C_);
      asm volatile("global_load_async_to_lds_b32 %0, %1, off"
                   :: "v"(dst), "v"(src)
                   : "memory");
    }
  }

  const int n = nr * nc;                 // >= 1 (spans >= 7 guaranteed by fix_span)
  float m = -3.402823466e+38f;
  int i = 0;

  // Stage 1: reduce positions guaranteed resident once ASYNCcnt <= 32.
  if (n > 32) {
    wait_asynccnt_32();                  // first n-32 copies retired (in-order)
    const int lim = n - 32;
    for (; i < lim; ++i) m = fmaxf(m, tile[i * C_ + c]);
  }
  // Stage 2: drain the tail. Only same-wave data is consumed (wave owns its
  // lanes' channels end-to-end), so no workgroup barrier is required.
  wait_asynccnt_0();
  for (; i < n; ++i) m = fmaxf(m, tile[i * C_ + c]);

  out[(((b * NREG_ + k) * P_ + ph) * P_ + pw) * C_ + c] = m;
}

// ---------------------------------------------------------------------------
// Launch: inputs are {features: (4,128,128,64) f32, roi: (4,160,4) f32};
// output is (4,32,7,7,64) f32 = 401408 elements. Workspace holds the
// 4*32 clipped int4 boxes (2 KB).
// ---------------------------------------------------------------------------
extern "C" void kernel_launch(void* const* d_in, const int* in_sizes, int n_in,
                              void* d_out, int out_size, void* d_ws, size_t ws_size,
                              hipStream_t stream) {
  (void)in_sizes; (void)n_in; (void)out_size; (void)ws_size;
  const float* feat = (const float*)d_in[0];
  const float* roi  = (const float*)d_in[1];
  float* out  = (float*)d_out;
  int* boxes  = (int*)d_ws;

  nms_clip_kernel<<<B_, R_, 0, stream>>>(roi, boxes);
  roi_pool_kernel<<<B_ * NREG_ * P_ * P_, C_, 0, stream>>>(feat, boxes, out);
}